// MoE_HyperModel_16930761081497
// MI455X (gfx1250) — compile-verified
//
#include <hip/hip_runtime.h>
#include <hip/hip_bf16.h>

typedef __attribute__((ext_vector_type(2))) float v2f;
typedef __attribute__((ext_vector_type(8))) float v8f;
typedef int v4i __attribute__((vector_size(16)));   // matches async builtin param pointee
typedef __attribute__((address_space(1))) v4i* gptr_v4i;
typedef __attribute__((address_space(3))) v4i* lptr_v4i;

#define EMB   1024
#define CTX   32
#define NEXP  16
#define H1    2048
#define H2    512
#define BATCH 8
#define NOUT  16384   // F_IN*R == R*F_OUT
#define MTOT  256     // CTX*BATCH

// Block tile: 64(M) x 128(N), BK=64. 8 waves in 2x4; each wave owns a 32x32 tile
// (4 v8f accumulators -> 4 independent WMMA chains, 2 A-frags + 2 B-frags per 4 wmma).
#define LDA  68    // As stride (64 rows x 64 cols): 4 mod 64 banks -> conflict-free col reads
#define LDB2 144   // Bs stride (64 rows x 128 cols): 16 mod 64 banks -> split-lane rows disjoint

#if defined(__has_builtin)
#  if __has_builtin(__builtin_amdgcn_global_load_async_to_lds_b128)
#    define HAVE_ASYNC_LDS 1
#  endif
#  if __has_builtin(__builtin_amdgcn_s_wait_asynccnt)
#    define HAVE_WAIT_ASYNC 1
#  endif
#endif

__device__ __forceinline__ void copy16(float* dst_lds, const float* src_gbl) {
#if defined(HAVE_ASYNC_LDS)
    __builtin_amdgcn_global_load_async_to_lds_b128(
        (gptr_v4i)(uintptr_t)src_gbl,
        (lptr_v4i)dst_lds,
        0, 0);
#else
    *reinterpret_cast<float4*>(dst_lds) = *reinterpret_cast<const float4*>(src_gbl);
#endif
}

__device__ __forceinline__ void wait_async() {
#if defined(HAVE_ASYNC_LDS)
#  if defined(HAVE_WAIT_ASYNC)
    __builtin_amdgcn_s_wait_asynccnt(0);
#  else
    asm volatile("s_wait_asynccnt 0x0" ::: "memory");
#  endif
#endif
}

// One BK=64 step for a 32x32 wave tile: 64 x v_wmma_f32_16x16x4_f32,
// A frags reused across N-subtiles, B frags reused across M-subtiles.
__device__ __forceinline__ void wmma_step(const float* __restrict__ As,
                                          const float* __restrict__ Bs,
                                          v8f& acc00, v8f& acc01, v8f& acc10, v8f& acc11,
                                          int wm, int wn, int lane) {
    const int half = lane >> 4;      // frag layout: lanes 0-15 hold K{0,1}, 16-31 hold K{2,3}
    const int r    = lane & 15;
    const float* arow0 = As + (wm * 32 + r) * LDA;
    const float* arow1 = arow0 + 16 * LDA;
    const float* bp0   = Bs + wn * 32 + r;
    const float* bp1   = bp0 + 16;
#pragma unroll
    for (int kk = 0; kk < 64; kk += 4) {
        const int ka = kk + half * 2;
        v2f a0, a1, b0, b1;
        a0.x = arow0[ka];        a0.y = arow0[ka + 1];
        a1.x = arow1[ka];        a1.y = arow1[ka + 1];
        b0.x = bp0[ka * LDB2];   b0.y = bp0[(ka + 1) * LDB2];
        b1.x = bp1[ka * LDB2];   b1.y = bp1[(ka + 1) * LDB2];
        acc00 = __builtin_amdgcn_wmma_f32_16x16x4_f32(false, a0, false, b0,
                                                      (short)0, acc00, false, false);
        acc01 = __builtin_amdgcn_wmma_f32_16x16x4_f32(false, a0, false, b1,
                                                      (short)0, acc01, false, false);
        acc10 = __builtin_amdgcn_wmma_f32_16x16x4_f32(false, a1, false, b0,
                                                      (short)0, acc10, false, false);
        acc11 = __builtin_amdgcn_wmma_f32_16x16x4_f32(false, a1, false, b1,
                                                      (short)0, acc11, false, false);
    }
}

// ---- gate1: g1[b,h] = relu(s @ Wg1 + bg1)  (memory bound; Wg1 streamed NT)
__global__ void __launch_bounds__(256) gate1_kernel(const float* __restrict__ s,
                                                    const float* __restrict__ Wg1,
                                                    const float* __restrict__ bg1,
                                                    float* __restrict__ g1) {
    __shared__ float sS[BATCH * EMB];
    const int h = blockIdx.x * 256 + threadIdx.x;
    for (int idx = threadIdx.x; idx < BATCH * EMB; idx += 256) sS[idx] = s[idx];
    __syncthreads();
    float acc[BATCH] = {};
#pragma unroll 4
    for (int i = 0; i < EMB; ++i) {
        const float w = __builtin_nontemporal_load(&Wg1[(size_t)i * H1 + h]);
#pragma unroll
        for (int b = 0; b < BATCH; ++b) acc[b] += sS[b * EMB + i] * w;
    }
#pragma unroll
    for (int b = 0; b < BATCH; ++b) {
        float v = acc[b] + bg1[h];
        g1[b * H1 + h] = v > 0.f ? v : 0.f;
    }
}

// ---- gate2 + softmax: weights[b,e]
__global__ void __launch_bounds__(128) gate2_kernel(const float* __restrict__ g1,
                                                    const float* __restrict__ Wg2,
                                                    const float* __restrict__ bg2,
                                                    float* __restrict__ wgt) {
    __shared__ float lg[BATCH * NEXP];
    const int b = threadIdx.x >> 4;
    const int e = threadIdx.x & 15;
    float acc = bg2[e];
    for (int j = 0; j < H1; ++j) acc += g1[b * H1 + j] * Wg2[j * NEXP + e];
    lg[threadIdx.x] = acc;
    __syncthreads();
    float mx = -3.4e38f;
#pragma unroll
    for (int k = 0; k < NEXP; ++k) mx = fmaxf(mx, lg[b * NEXP + k]);
    float sum = 0.f;
#pragma unroll
    for (int k = 0; k < NEXP; ++k) sum += __expf(lg[b * NEXP + k] - mx);
    wgt[threadIdx.x] = __expf(acc - mx) / sum;
}

// ---- base[e,b,h] = s @ W1[e,:EMB,:] + b1[e]  (memory bound on 134 MB W1 -> NT loads)
__global__ void __launch_bounds__(256) base_kernel(const float* __restrict__ s,
                                                   const float* __restrict__ W1,
                                                   const float* __restrict__ b1,
                                                   float* __restrict__ base) {
    __shared__ float sS[BATCH * EMB];
    const int e = blockIdx.y;
    const int h = blockIdx.x * 256 + threadIdx.x;
    for (int idx = threadIdx.x; idx < BATCH * EMB; idx += 256) sS[idx] = s[idx];
    __syncthreads();
    float acc[BATCH] = {};
    const float* W1e = W1 + (size_t)e * (EMB + CTX) * H1;
#pragma unroll 4
    for (int i = 0; i < EMB; ++i) {
        const float w = __builtin_nontemporal_load(&W1e[(size_t)i * H1 + h]);
#pragma unroll
        for (int b = 0; b < BATCH; ++b) acc[b] += sS[b * EMB + i] * w;
    }
    const float bias = b1[e * H1 + h];
#pragma unroll
    for (int b = 0; b < BATCH; ++b)
        base[(size_t)(e * BATCH + b) * H1 + h] = acc[b] + bias;
}

// ---- tile loaders -----------------------------------------------------------
__device__ __forceinline__ void zg_load(float* __restrict__ AsN, float* __restrict__ BsN,
                                        const float* __restrict__ basee,
                                        const float* __restrict__ W1ce,
                                        const float* __restrict__ W2e,
                                        const float* __restrict__ wgt,
                                        int e, int m0, int n0, int k0, int tid) {
    // B tile: 64 x 128 slab of W2[e]   (async global->LDS)
    for (int t = tid; t < 64 * 32; t += 256) {
        const int rr = t >> 5, c4 = (t & 31) << 2;
        copy16(&BsN[rr * LDB2 + c4], &W2e[(size_t)(k0 + rr) * H2 + n0 + c4]);
    }
    // A tile: 64 x 64 fused w[b,e]*relu(base + W1c), computed while cur buf runs WMMA
    for (int t = tid; t < 64 * 64; t += 256) {
        const int row = t >> 6, cc = t & 63;
        const int mg = m0 + row, c = mg >> 3, b = mg & 7;
        float v = basee[b * H1 + k0 + cc] + W1ce[(size_t)c * H1 + k0 + cc];
        v = v > 0.f ? v : 0.f;
        AsN[row * LDA + cc] = wgt[b * NEXP + e] * v;
    }
}

__device__ __forceinline__ void og_load(float* __restrict__ AsN, float* __restrict__ BsN,
                                        const float* __restrict__ z,
                                        const float* __restrict__ W,
                                        int m0, int n0, int k0, int tid) {
    for (int t = tid; t < 64 * 32; t += 256) {
        const int rr = t >> 5, c4 = (t & 31) << 2;
        copy16(&BsN[rr * LDB2 + c4], &W[(size_t)(k0 + rr) * NOUT + n0 + c4]);
    }
    for (int t = tid; t < 64 * 16; t += 256) {
        const int rr = t >> 4, c4 = (t & 15) << 2;
        copy16(&AsN[rr * LDA + c4], &z[(size_t)(m0 + rr) * H2 + k0 + c4]);
    }
}

// ---- per-expert WMMA GEMM: zpart[e] = A(e) @ W2[e], double-buffered pipeline
__global__ void __launch_bounds__(256) z_gemm_kernel(const float* __restrict__ base,
                                                     const float* __restrict__ W1,
                                                     const float* __restrict__ W2,
                                                     const float* __restrict__ wgt,
                                                     float* __restrict__ zpart) {
    __shared__ float As[2][64 * LDA];
    __shared__ float Bs[2][64 * LDB2];
    const int n0 = blockIdx.x * 128;  // 4 blocks over H2
    const int m0 = blockIdx.y * 64;   // 4 blocks over M
    const int e  = blockIdx.z;
    const int tid = threadIdx.x, lane = tid & 31, w = tid >> 5;
    const int wm = w >> 2, wn = w & 3;
    const float* W2e   = W2 + (size_t)e * H1 * H2;
    const float* W1ce  = W1 + ((size_t)e * (EMB + CTX) + EMB) * H1;
    const float* basee = base + (size_t)e * BATCH * H1;
    v8f acc00 = {}, acc01 = {}, acc10 = {}, acc11 = {};

    zg_load(As[0], Bs[0], basee, W1ce, W2e, wgt, e, m0, n0, 0, tid);
    wait_async();
    __syncthreads();
    int cur = 0;
    for (int k0 = 0; k0 < H1; k0 += 64) {
        if (k0 + 64 < H1)
            zg_load(As[cur ^ 1], Bs[cur ^ 1], basee, W1ce, W2e, wgt, e, m0, n0, k0 + 64, tid);
        wmma_step(As[cur], Bs[cur], acc00, acc01, acc10, acc11, wm, wn, lane);
        wait_async();
        __syncthreads();
        cur ^= 1;
    }

    const int half = lane >> 4, r = lane & 15;
    float* outp = zpart + (size_t)e * MTOT * H2;
#pragma unroll
    for (int v = 0; v < 8; ++v) {
        const int m = m0 + wm * 32 + v + half * 8;   // C/D layout: VGPR v -> rows v, v+8
        const int n = n0 + wn * 32 + r;
        outp[(size_t)m * H2 + n]             = acc00[v];
        outp[(size_t)m * H2 + n + 16]        = acc01[v];
        outp[(size_t)(m + 16) * H2 + n]      = acc10[v];
        outp[(size_t)(m + 16) * H2 + n + 16] = acc11[v];
    }
}

// ---- z[m,d] = sum_e zpart[e,m,d] + sum_e w[b,e]*b2[e,d]
__global__ void __launch_bounds__(256) z_reduce_kernel(const float* __restrict__ zpart,
                                                       const float* __restrict__ wgt,
                                                       const float* __restrict__ b2,
                                                       float* __restrict__ z) {
    const int idx = blockIdx.x * 256 + threadIdx.x;
    const int m = idx >> 9, d = idx & 511, b = m & 7;
    float sum = 0.f;
#pragma unroll
    for (int e = 0; e < NEXP; ++e)
        sum += zpart[(size_t)e * MTOT * H2 + idx] + wgt[b * NEXP + e] * b2[e * H2 + d];
    z[idx] = sum;
}

// ---- out = z @ W + bias : M=256, N=16384, K=512 -> NT stores straight to d_out
__global__ void __launch_bounds__(256) out_gemm_kernel(const float* __restrict__ z,
                                                       const float* __restrict__ W,
                                                       const float* __restrict__ bias,
                                                       float* __restrict__ out) {
    __shared__ float As[2][64 * LDA];
    __shared__ float Bs[2][64 * LDB2];
    const int n0 = blockIdx.x * 128;  // 128 blocks over N
    const int m0 = blockIdx.y * 64;   // 4 blocks over M
    const int tid = threadIdx.x, lane = tid & 31, w = tid >> 5;
    const int wm = w >> 2, wn = w & 3;
    v8f acc00 = {}, acc01 = {}, acc10 = {}, acc11 = {};

    og_load(As[0], Bs[0], z, W, m0, n0, 0, tid);
    wait_async();
    __syncthreads();
    int cur = 0;
    for (int k0 = 0; k0 < H2; k0 += 64) {
        if (k0 + 64 < H2)
            og_load(As[cur ^ 1], Bs[cur ^ 1], z, W, m0, n0, k0 + 64, tid);
        wmma_step(As[cur], Bs[cur], acc00, acc01, acc10, acc11, wm, wn, lane);
        wait_async();
        __syncthreads();
        cur ^= 1;
    }

    const int half = lane >> 4, r = lane & 15;
#pragma unroll
    for (int v = 0; v < 8; ++v) {
        const int m = m0 + wm * 32 + v + half * 8;
        const int n = n0 + wn * 32 + r;
        __builtin_nontemporal_store(acc00[v] + bias[n],      &out[(size_t)m * NOUT + n]);
        __builtin_nontemporal_store(acc01[v] + bias[n + 16], &out[(size_t)m * NOUT + n + 16]);
        __builtin_nontemporal_store(acc10[v] + bias[n],      &out[(size_t)(m + 16) * NOUT + n]);
        __builtin_nontemporal_store(acc11[v] + bias[n + 16], &out[(size_t)(m + 16) * NOUT + n + 16]);
    }
}

extern "C" void kernel_launch(void* const* d_in, const int* in_sizes, int n_in,
                              void* d_out, int out_size, void* d_ws, size_t ws_size,
                              hipStream_t stream) {
    const float* s   = (const float*)d_in[0];
    const float* Wg1 = (const float*)d_in[1];
    const float* bg1 = (const float*)d_in[2];
    const float* Wg2 = (const float*)d_in[3];
    const float* bg2 = (const float*)d_in[4];
    const float* W1  = (const float*)d_in[5];
    const float* b1  = (const float*)d_in[6];
    const float* W2  = (const float*)d_in[7];
    const float* b2  = (const float*)d_in[8];
    const float* WA  = (const float*)d_in[9];
    const float* bA  = (const float*)d_in[10];
    const float* WB  = (const float*)d_in[11];
    const float* bB  = (const float*)d_in[12];

    float* ws    = (float*)d_ws;
    float* g1    = ws;                               // 16384
    float* wgt   = g1 + BATCH * H1;                  // 128
    float* base  = wgt + BATCH * NEXP;               // 262144
    float* zpart = base + NEXP * BATCH * H1;         // 2097152
    float* z     = zpart + (size_t)NEXP * MTOT * H2; // 131072

    float* outA = (float*)d_out;
    float* outB = outA + (size_t)MTOT * NOUT;

    gate1_kernel<<<H1 / 256, 256, 0, stream>>>(s, Wg1, bg1, g1);
    gate2_kernel<<<1, 128, 0, stream>>>(g1, Wg2, bg2, wgt);
    base_kernel<<<dim3(H1 / 256, NEXP), 256, 0, stream>>>(s, W1, b1, base);
    z_gemm_kernel<<<dim3(H2 / 128, MTOT / 64, NEXP), 256, 0, stream>>>(base, W1, W2, wgt, zpart);
    z_reduce_kernel<<<(MTOT * H2) / 256, 256, 0, stream>>>(zpart, wgt, b2, z);
    out_gemm_kernel<<<dim3(NOUT / 128, MTOT / 64), 256, 0, stream>>>(z, WA, bA, outA);
    out_gemm_kernel<<<dim3(NOUT / 128, MTOT / 64), 256, 0, stream>>>(z, WB, bB, outB);
}